// DiGCN_IB_Sum_24318104830208
// MI455X (gfx1250) — compile-verified
//
#include <hip/hip_runtime.h>
#include <hip/hip_bf16.h>

// DiGCN inception-block stack for MI455X (gfx1250, wave32).
//   per layer: x0 = x@lnW + (lnb+c1b+c2b); h1 = x@c1W; h2 = x@c2W   (fused WMMA GEMM)
//              x  = x0 + scatter_add(w1*h1[src1] -> dst1) + scatter_add(w2*h2[src2] -> dst2)
// GEMMs use V_WMMA_F32_16X16X4_F32 (exact fp32). Scatters use f32 L2 atomics.

#define FDIM 128
#define LPAD 132   // LDS row stride (floats) to avoid bank conflicts

typedef float v2f __attribute__((ext_vector_type(2)));
typedef float v8f __attribute__((ext_vector_type(8)));

// One block = 256 threads = 8 waves; block handles a 16-row tile of x.
// Wave w owns output column tile [16w, 16w+16) and loops over the 3 weight mats.
__global__ __launch_bounds__(256)
void gemm3_wmma_kernel(const float* __restrict__ xin,
                       const float* __restrict__ Wln,
                       const float* __restrict__ W1,
                       const float* __restrict__ W2,
                       const float* __restrict__ bln,
                       const float* __restrict__ b1,
                       const float* __restrict__ b2,
                       float* __restrict__ x0out,   // gets x@lnW + (lnb+b1+b2)
                       float* __restrict__ h1out,   // gets x@c1W
                       float* __restrict__ h2out,   // gets x@c2W
                       int nrows)
{
    __shared__ float sx[16 * LPAD];

    const int t = threadIdx.x;
    const int rowbase = blockIdx.x * 16;

    // Cooperative load of 16x128 tile of x into LDS (coalesced, clamp tail rows).
    #pragma unroll
    for (int i = 0; i < 8; ++i) {
        int flat = t + i * 256;          // 0..2047
        int r = flat >> 7;               // 0..15
        int c = flat & (FDIM - 1);       // 0..127
        int gr = rowbase + r;
        if (gr >= nrows) gr = nrows - 1; // clamp (value irrelevant, store guarded)
        sx[r * LPAD + c] = xin[(size_t)gr * FDIM + c];
    }
    __syncthreads();

    const int lane = t & 31;
    const int wave = t >> 5;       // 0..7 -> N tile
    const int half = lane >> 4;    // 0 or 1
    const int lm   = lane & 15;    // 0..15
    const int koff = half * 2;     // K sub-offset for A/B fragments
    const int ncol = wave * 16 + lm;

    const float* Ws[3]   = {Wln, W1, W2};
    float*       outs[3] = {x0out, h1out, h2out};
    const float  biasSum = bln[ncol] + b1[ncol] + b2[ncol];
    const bool   fullTile = (rowbase + 16 <= nrows);   // block-uniform

    #pragma unroll
    for (int j = 0; j < 3; ++j) {
        const float* __restrict__ W = Ws[j];
        v8f acc = {0.f, 0.f, 0.f, 0.f, 0.f, 0.f, 0.f, 0.f};

        // K = 128, 4 per WMMA -> 32 MACs. EXEC is all-1s here (no divergence).
        #pragma unroll 4
        for (int kk = 0; kk < FDIM; kk += 4) {
            v2f a, b;
            // A (16x4): lane {half,lm}: VGPR0 = A[lm][kk+2*half], VGPR1 = +1
            a.x = sx[lm * LPAD + kk + koff];
            a.y = sx[lm * LPAD + kk + koff + 1];
            // B (4x16): lane {half,lm}: VGPR0 = W[kk+2*half][ncol], VGPR1 = +1 row
            b.x = W[(kk + koff) * FDIM + ncol];
            b.y = W[(kk + koff + 1) * FDIM + ncol];
            acc = __builtin_amdgcn_wmma_f32_16x16x4_f32(
                      /*neg_a=*/false, a, /*neg_b=*/false, b,
                      /*c_mod=*/(short)0, acc,
                      /*reuse_a=*/false, /*reuse_b=*/false);
        }

        const float badd = (j == 0) ? biasSum : 0.0f;
        // D layout: VGPR v holds M = v + 8*half, N = ncol
        float* __restrict__ p =
            outs[j] + (size_t)(rowbase + half * 8) * FDIM + ncol;
        if (fullTile) {
            // uniform fast path: no exec-mask churn, immediate-offset stores
            #pragma unroll
            for (int v = 0; v < 8; ++v)
                p[(size_t)v * FDIM] = acc[v] + badd;
        } else {
            #pragma unroll
            for (int v = 0; v < 8; ++v) {
                if (rowbase + half * 8 + v < nrows)
                    p[(size_t)v * FDIM] = acc[v] + badd;
            }
        }
    }
}

// out[dst[e]] += w[e] * h[src[e]]  ; one thread per (edge, 4-float chunk)
__global__ __launch_bounds__(256)
void scatter_add_kernel(const float* __restrict__ h,
                        const float* __restrict__ w,
                        const int* __restrict__ src,
                        const int* __restrict__ dst,
                        float* __restrict__ out,
                        int ne)
{
    long long tid = (long long)blockIdx.x * blockDim.x + threadIdx.x;
    int e = (int)(tid >> 5);
    if (e >= ne) return;
    int f = (int)(tid & 31) << 2;          // feature offset, 16B aligned

    const float ww = w[e];
    const int   s  = src[e];
    const int   d  = dst[e];

    const float4 hv = *(const float4*)(h + (size_t)s * FDIM + f);
    float* op = out + (size_t)d * FDIM + f;
    unsafeAtomicAdd(op + 0, ww * hv.x);
    unsafeAtomicAdd(op + 1, ww * hv.y);
    unsafeAtomicAdd(op + 2, ww * hv.z);
    unsafeAtomicAdd(op + 3, ww * hv.w);
}

extern "C" void kernel_launch(void* const* d_in, const int* in_sizes, int n_in,
                              void* d_out, int out_size, void* d_ws, size_t ws_size,
                              hipStream_t stream) {
    // setup_inputs() order:
    const float* x   = (const float*)d_in[0];   // [N,128]
    const float* ea1 = (const float*)d_in[1];   // [E]
    const float* ea2 = (const float*)d_in[2];   // [E]
    const float* lnW = (const float*)d_in[3];   // [3,128,128]
    const float* lnb = (const float*)d_in[4];   // [3,128]
    const float* c1W = (const float*)d_in[5];
    const float* c1b = (const float*)d_in[6];
    const float* c2W = (const float*)d_in[7];
    const float* c2b = (const float*)d_in[8];
    const int*   ei1 = (const int*)d_in[9];     // [2,E]
    const int*   ei2 = (const int*)d_in[10];    // [2,E]
    // d_in[11] = batch (unused: sorted-batch concat == identity)

    const int N = in_sizes[0] / FDIM;
    const int E = in_sizes[1];

    float* ws   = (float*)d_ws;
    size_t nf   = (size_t)N * FDIM;
    float* bufA = ws;            // layer-0 output
    float* bufB = ws + nf;       // layer-1 output
    float* h1   = ws + 2 * nf;   // x @ c1W scratch
    float* h2   = ws + 3 * nf;   // x @ c2W scratch
    float* out  = (float*)d_out;

    const int gemmBlocks = (N + 15) / 16;
    const int scatBlocks = (int)(((long long)E * 32 + 255) / 256);

    const float* xin = x;
    for (int l = 0; l < 3; ++l) {
        float* xout = (l == 2) ? out : ((l == 0) ? bufA : bufB);
        const size_t wOff = (size_t)l * FDIM * FDIM;
        const size_t bOff = (size_t)l * FDIM;

        gemm3_wmma_kernel<<<gemmBlocks, 256, 0, stream>>>(
            xin, lnW + wOff, c1W + wOff, c2W + wOff,
            lnb + bOff, c1b + bOff, c2b + bOff,
            xout, h1, h2, N);

        scatter_add_kernel<<<scatBlocks, 256, 0, stream>>>(
            h1, ea1, ei1, ei1 + E, xout, E);
        scatter_add_kernel<<<scatBlocks, 256, 0, stream>>>(
            h2, ea2, ei2, ei2 + E, xout, E);

        xin = xout;
    }
}